// SlotAttention_35725537968195
// MI455X (gfx1250) — compile-verified
//
#include <hip/hip_runtime.h>
#include <math.h>

#define B_ 16
#define T_ 4096
#define D_ 128
#define K_ 8
#define N_ITERS_ 3

typedef float v2f __attribute__((ext_vector_type(2)));
typedef float v8f __attribute__((ext_vector_type(8)));

// D = A(16x4) * B(4x16) + C, f32 everywhere -> v_wmma_f32_16x16x4_f32
static __device__ __forceinline__ v8f wmma_f32_16x16x4(v2f a, v2f b, v8f c) {
  return __builtin_amdgcn_wmma_f32_16x16x4_f32(false, a, false, b, (short)0, c,
                                               false, false);
}

static __device__ __forceinline__ float softplusf_(float x) {
  return (x > 20.f) ? x : log1pf(__expf(x));
}
static __device__ __forceinline__ float sigmoidf_(float x) {
  return 1.f / (1.f + __expf(-x));
}

// ---------------------------------------------------------------------------
// slots init: broadcast slot_mu over batch
// ---------------------------------------------------------------------------
__global__ void init_slots_kernel(const float* __restrict__ slot_mu,
                                  float* __restrict__ slots_ws) {
  const int i = blockIdx.x * blockDim.x + threadIdx.x;
  if (i < B_ * K_ * D_) slots_ws[i] = slot_mu[i % (K_ * D_)];
}

// ---------------------------------------------------------------------------
// Kernel 1: x_n = LN(x); k = x_n @ Wk^T; v = x_n @ Wv^T   (f32 WMMA)
// 4 waves/block, each wave owns 16 rows of T.  Lane caches half of its row
// (the e-positions its A-fragments need) in 64 VGPRs; LN stats combined with
// the partner lane via shfl_xor(16).
// ---------------------------------------------------------------------------
__global__ __launch_bounds__(128) void ln_kv_kernel(
    const float* __restrict__ x, const float* __restrict__ Wk,
    const float* __restrict__ Wv, const float* __restrict__ lnw,
    const float* __restrict__ lnb, float* __restrict__ k_ws,
    float* __restrict__ v_ws) {
  const int wave = threadIdx.x >> 5, lane = threadIdx.x & 31;
  const int half = lane >> 4, nl = lane & 15;
  const int b = blockIdx.x / (T_ / 64);
  const int t0 = (blockIdx.x % (T_ / 64)) * 64 + wave * 16;
  const float* xrow = x + ((size_t)b * T_ + t0 + nl) * D_ + 2 * half;

  float xa[64];
  float s = 0.f, s2 = 0.f;
#pragma unroll
  for (int kk = 0; kk < 32; ++kk) {
    float2 t = *(const float2*)(xrow + kk * 4);
    xa[2 * kk] = t.x;
    xa[2 * kk + 1] = t.y;
    s += t.x + t.y;
    s2 += t.x * t.x + t.y * t.y;
  }
  s += __shfl_xor(s, 16, 32);
  s2 += __shfl_xor(s2, 16, 32);
  const float mu = s * (1.f / 128.f);
  const float var = s2 * (1.f / 128.f) - mu * mu;
  const float rstd = rsqrtf(var + 1e-5f);
#pragma unroll
  for (int kk = 0; kk < 32; ++kk) {
    const int e = kk * 4 + 2 * half;
    float2 w = *(const float2*)(lnw + e);
    float2 bb = *(const float2*)(lnb + e);
    xa[2 * kk] = (xa[2 * kk] - mu) * rstd * w.x + bb.x;
    xa[2 * kk + 1] = (xa[2 * kk + 1] - mu) * rstd * w.y + bb.y;
  }

  const float* Ws[2] = {Wk, Wv};
  float* Os[2] = {k_ws, v_ws};
#pragma unroll
  for (int m = 0; m < 2; ++m) {
    const float* W = Ws[m];
    float* O = Os[m];
    for (int nt = 0; nt < 8; ++nt) {
      const int n = nt * 16 + nl;  // output column (B-fragment N index)
      v8f c = {};
#pragma unroll
      for (int kk = 0; kk < 32; ++kk) {
        const int e = kk * 4 + 2 * half;
        v2f a;
        a.x = xa[2 * kk];
        a.y = xa[2 * kk + 1];
        // B[e][n] = W[n][e]  (W is 64KB, L2/L0-resident, heavily reused)
        float2 wv = *(const float2*)(W + (size_t)n * D_ + e);
        v2f bf;
        bf.x = wv.x;
        bf.y = wv.y;
        c = wmma_f32_16x16x4(a, bf, c);
      }
#pragma unroll
      for (int i = 0; i < 8; ++i)
        O[((size_t)b * T_ + t0 + i + 8 * half) * D_ + n] = c[i];
    }
  }
}

// ---------------------------------------------------------------------------
// Kernel 2: q = LN(slots) @ Wq^T (tiny; VALU).  Pads q to 16 slot-rows with
// zeros so the dots GEMM uses one full 16-wide N tile.  Optionally copies the
// final slots to d_out.
// ---------------------------------------------------------------------------
static __device__ __forceinline__ float block_sum_128(float v, float* red) {
#pragma unroll
  for (int d = 16; d > 0; d >>= 1) v += __shfl_xor(v, d, 32);
  __syncthreads();
  if ((threadIdx.x & 31) == 0) red[threadIdx.x >> 5] = v;
  __syncthreads();
  return red[0] + red[1] + red[2] + red[3];
}

__global__ __launch_bounds__(128) void q_kernel(
    const float* __restrict__ slots_ws, const float* __restrict__ Wq,
    const float* __restrict__ lnw, const float* __restrict__ lnb,
    float* __restrict__ q_ws, float* __restrict__ slots_out) {
  __shared__ float sn[128];
  __shared__ float red[4];
  const int b = blockIdx.x, tid = threadIdx.x;
  for (int k = 0; k < K_; ++k) {
    const float sv = slots_ws[((size_t)b * K_ + k) * D_ + tid];
    const float mu = block_sum_128(sv, red) * (1.f / 128.f);
    const float dd = sv - mu;
    const float var = block_sum_128(dd * dd, red) * (1.f / 128.f);
    sn[tid] = dd * rsqrtf(var + 1e-5f) * lnw[tid] + lnb[tid];
    __syncthreads();
    float acc = 0.f;
    const float* wrow = Wq + (size_t)tid * D_;
    for (int e = 0; e < D_; ++e) acc += sn[e] * wrow[e];
    q_ws[(size_t)b * 2048 + k * D_ + tid] = acc;
    if (slots_out) slots_out[((size_t)b * K_ + k) * D_ + tid] = sv;
  }
  for (int k = K_; k < 16; ++k) q_ws[(size_t)b * 2048 + k * D_ + tid] = 0.f;
}

// ---------------------------------------------------------------------------
// Kernel 3: dotsT[t][k] = k_row(t) . q_row(k) * scale  (f32 WMMA, one N-tile)
// then softmax over the SLOT axis (k lives across lanes -> octet shfl_xor).
// Non-final: writes attn padded to 16 slots (zeros) for the updates GEMM.
// Final: writes softmax directly to d_out[(b,k,t)].
// k_ws (33.5 MB) is L2-resident -> this kernel runs out of L2.
// ---------------------------------------------------------------------------
__global__ __launch_bounds__(256) void dots_kernel(
    const float* __restrict__ q_ws, const float* __restrict__ k_ws,
    float* __restrict__ attn_ws, float* __restrict__ final_out) {
  const int wave = threadIdx.x >> 5, lane = threadIdx.x & 31;
  const int half = lane >> 4, nl = lane & 15;
  const int b = blockIdx.x >> 5;  // T_/128 == 32 tiles per batch
  const int t0 = (blockIdx.x & 31) * 128 + wave * 16;
  const float* krow = k_ws + ((size_t)b * T_ + t0 + nl) * D_ + 2 * half;
  const float* qrow = q_ws + (size_t)b * 2048 + (size_t)nl * D_ + 2 * half;

  float ka[64], qa[64];
#pragma unroll
  for (int kk = 0; kk < 32; ++kk) {
    float2 kv = *(const float2*)(krow + kk * 4);
    float2 qv = *(const float2*)(qrow + kk * 4);
    ka[2 * kk] = kv.x;
    ka[2 * kk + 1] = kv.y;
    qa[2 * kk] = qv.x;
    qa[2 * kk + 1] = qv.y;
  }
  v8f c = {};
#pragma unroll
  for (int kk = 0; kk < 32; ++kk) {
    v2f a;
    a.x = ka[2 * kk];
    a.y = ka[2 * kk + 1];
    v2f bf;
    bf.x = qa[2 * kk];
    bf.y = qa[2 * kk + 1];
    c = wmma_f32_16x16x4(a, bf, c);
  }

  const float scale = 0.088388347648318447f;  // 128^-0.5
  const bool valid = nl < K_;
#pragma unroll
  for (int i = 0; i < 8; ++i) {
    float vv = valid ? c[i] * scale : -3.0e38f;
    float mx = vv;
    mx = fmaxf(mx, __shfl_xor(mx, 1, 32));
    mx = fmaxf(mx, __shfl_xor(mx, 2, 32));
    mx = fmaxf(mx, __shfl_xor(mx, 4, 32));
    float ex = valid ? __expf(vv - mx) : 0.f;
    float sm = ex;
    sm += __shfl_xor(sm, 1, 32);
    sm += __shfl_xor(sm, 2, 32);
    sm += __shfl_xor(sm, 4, 32);
    const float p = valid ? ex / sm : 0.f;
    const int trow = t0 + i + 8 * half;
    if (final_out) {
      if (valid) final_out[((size_t)b * K_ + nl) * T_ + trow] = p;
    } else {
      attn_ws[((size_t)b * T_ + trow) * 16 + nl] = p;  // padded, coalesced
    }
  }
}

// ---------------------------------------------------------------------------
// Kernel 4: U = attn^T @ v (f32 WMMA, K-dim = T), per-slot T-sums, yield
// activation, GRU cell.  One block per batch; 8 waves split T; partials
// combined with LDS float atomics (16 KB LDS total).
// ---------------------------------------------------------------------------
__global__ __launch_bounds__(256) void updates_gru_kernel(
    const float* __restrict__ attn_ws, const float* __restrict__ v_ws,
    float* __restrict__ slots_ws, const float* __restrict__ sigma_y_raw,
    const float* __restrict__ p_raw, const float* __restrict__ W_ih,
    const float* __restrict__ W_hh, const float* __restrict__ b_ih,
    const float* __restrict__ b_hh) {
  __shared__ float U[16 * 128];
  __shared__ float upd[8 * 128];
  __shared__ float sl[8 * 128];
  __shared__ float Ssl[8];
  const int b = blockIdx.x, tid = threadIdx.x;
  const int wave = tid >> 5, lane = tid & 31;
  const int half = lane >> 4, nl = lane & 15;

  for (int i = tid; i < 2048; i += 256) U[i] = 0.f;
  for (int i = tid; i < 1024; i += 256) sl[i] = slots_ws[(size_t)b * 1024 + i];
  {  // deterministic per-slot attention row-sums (wave w owns slot w)
    float sacc = 0.f;
    for (int t = lane; t < T_; t += 32)
      sacc += attn_ws[((size_t)b * T_ + t) * 16 + wave];
#pragma unroll
    for (int d = 16; d > 0; d >>= 1) sacc += __shfl_xor(sacc, d, 32);
    if (lane == 0) Ssl[wave] = sacc;
  }
  __syncthreads();

  const int tbase = wave * 512;  // each wave reduces 512 of the 4096 t's
#pragma unroll 1
  for (int nt = 0; nt < 8; ++nt) {
    v8f c = {};
    for (int tk = 0; tk < 128; ++tk) {
      const int t = tbase + tk * 4 + 2 * half;
      v2f a;  // A[m=slot][kdim=t] = attn[b, slot, t]
      a.x = attn_ws[((size_t)b * T_ + t) * 16 + nl];
      a.y = attn_ws[((size_t)b * T_ + t + 1) * 16 + nl];
      v2f bf;  // B[t][n=d] = v[b, t, d]  (coalesced, L2-resident)
      bf.x = v_ws[((size_t)b * T_ + t) * D_ + nt * 16 + nl];
      bf.y = v_ws[((size_t)b * T_ + t + 1) * D_ + nt * 16 + nl];
      c = wmma_f32_16x16x4(a, bf, c);
    }
#pragma unroll
    for (int i = 0; i < 8; ++i)
      atomicAdd(&U[(i + 8 * half) * 128 + nt * 16 + nl], c[i]);
  }
  __syncthreads();

  // updates = (U / (sum+eps)) passed through the yield activation
  for (int idx = tid; idx < 1024; idx += 256) {
    const int m = idx >> 7, d = idx & 127;
    float u = U[m * 128 + d] / (Ssl[m] + 1e-8f);
    const float sy = softplusf_(sigma_y_raw[m * 128 + d]) + 0.01f;
    const float pw = 1.5f + softplusf_(p_raw[m]);
    const float ratio = fminf(fabsf(u / sy), 15.f);
    u = u / powf(1.f + powf(ratio, pw), 1.f / pw);
    upd[idx] = u;
  }
  __syncthreads();

  // GRU cell: input = upd, hidden = sl
  for (int idx = tid; idx < 1024; idx += 256) {
    const int m = idx >> 7, d = idx & 127;
    float ri = b_ih[d], zi = b_ih[128 + d], ni = b_ih[256 + d];
    float rh = b_hh[d], zh = b_hh[128 + d], nh = b_hh[256 + d];
    const float* u_m = upd + m * 128;
    const float* s_m = sl + m * 128;
    const float* wi_r = W_ih + (size_t)d * 128;
    const float* wi_z = W_ih + (size_t)(128 + d) * 128;
    const float* wi_n = W_ih + (size_t)(256 + d) * 128;
    const float* wh_r = W_hh + (size_t)d * 128;
    const float* wh_z = W_hh + (size_t)(128 + d) * 128;
    const float* wh_n = W_hh + (size_t)(256 + d) * 128;
    for (int e = 0; e < 128; ++e) {
      const float ue = u_m[e], se = s_m[e];
      ri += ue * wi_r[e];
      zi += ue * wi_z[e];
      ni += ue * wi_n[e];
      rh += se * wh_r[e];
      zh += se * wh_z[e];
      nh += se * wh_n[e];
    }
    const float r = sigmoidf_(ri + rh);
    const float z = sigmoidf_(zi + zh);
    const float n = tanhf(ni + r * nh);
    slots_ws[(size_t)b * 1024 + idx] = (1.f - z) * n + z * s_m[d];
  }
}

// ---------------------------------------------------------------------------
extern "C" void kernel_launch(void* const* d_in, const int* in_sizes, int n_in,
                              void* d_out, int out_size, void* d_ws,
                              size_t ws_size, hipStream_t stream) {
  (void)in_sizes; (void)n_in; (void)out_size; (void)ws_size;
  const float* x       = (const float*)d_in[0];
  const float* slot_mu = (const float*)d_in[1];
  const float* Wq      = (const float*)d_in[2];
  const float* Wk      = (const float*)d_in[3];
  const float* Wv      = (const float*)d_in[4];
  const float* W_ih    = (const float*)d_in[5];
  const float* W_hh    = (const float*)d_in[6];
  const float* b_ih    = (const float*)d_in[7];
  const float* b_hh    = (const float*)d_in[8];
  const float* ln_s_w  = (const float*)d_in[9];
  const float* ln_s_b  = (const float*)d_in[10];
  const float* ln_x_w  = (const float*)d_in[11];
  const float* ln_x_b  = (const float*)d_in[12];
  const float* sigma_y = (const float*)d_in[13];
  const float* p_raw   = (const float*)d_in[14];

  float* ws = (float*)d_ws;
  float* k_ws = ws;                                   // B*T*D
  float* v_ws = k_ws + (size_t)B_ * T_ * D_;          // B*T*D
  float* attn_ws = v_ws + (size_t)B_ * T_ * D_;       // B*T*16 (padded slots)
  float* q_ws = attn_ws + (size_t)B_ * T_ * 16;       // B*16*D (padded slots)
  float* slots_ws = q_ws + (size_t)B_ * 16 * D_;      // B*K*D

  float* out_slots = (float*)d_out;
  float* out_attn = out_slots + (size_t)B_ * K_ * D_;

  init_slots_kernel<<<(B_ * K_ * D_) / 256, 256, 0, stream>>>(slot_mu,
                                                              slots_ws);
  ln_kv_kernel<<<B_ * (T_ / 64), 128, 0, stream>>>(x, Wk, Wv, ln_x_w, ln_x_b,
                                                   k_ws, v_ws);
  for (int it = 0; it < N_ITERS_; ++it) {
    q_kernel<<<B_, 128, 0, stream>>>(slots_ws, Wq, ln_s_w, ln_s_b, q_ws,
                                     (float*)nullptr);
    dots_kernel<<<B_ * (T_ / 128), 256, 0, stream>>>(q_ws, k_ws, attn_ws,
                                                     (float*)nullptr);
    updates_gru_kernel<<<B_, 256, 0, stream>>>(attn_ws, v_ws, slots_ws,
                                               sigma_y, p_raw, W_ih, W_hh,
                                               b_ih, b_hh);
  }
  q_kernel<<<B_, 128, 0, stream>>>(slots_ws, Wq, ln_s_w, ln_s_b, q_ws,
                                   out_slots);
  dots_kernel<<<B_ * (T_ / 128), 256, 0, stream>>>(q_ws, k_ws, attn_ws,
                                                   out_attn);
}